// Biholomorphic_k3_45921790329589
// MI455X (gfx1250) — compile-verified
//
#include <hip/hip_runtime.h>

typedef float v2f __attribute__((ext_vector_type(2)));
typedef float v8f __attribute__((ext_vector_type(8)));

// Monomial index triples (i<=j<=k) over D=5, in reference (TI,TJ,TK) order.
__constant__ unsigned char c_TI[35] = {0,0,0,0,0,0,0,0,0,0,0,0,0,0,0,1,1,1,1,1,1,1,1,1,1,2,2,2,2,2,2,3,3,3,4};
__constant__ unsigned char c_TJ[35] = {0,0,0,0,0,1,1,1,1,2,2,2,3,3,4,1,1,1,1,2,2,2,3,3,4,2,2,2,3,3,4,3,3,4,4};
__constant__ unsigned char c_TK[35] = {0,1,2,3,4,1,2,3,4,2,3,4,3,4,4,1,2,3,4,2,3,4,3,4,4,2,3,4,3,4,4,3,4,4,4};

#define WAVES_PER_BLOCK 8
// Per-wave LDS slice layout (floats):
//  [  0.. 47] zr (padded, idx>=35 are 0)
//  [ 48.. 95] zi (padded)
//  [ 96..111] xr[5] at +0, xi[5] at +8 (dynamic-index staging)
//  [112..1343] packed output row (1225 used; word 1228 = scatter dummy slot)
#define SLICE   1344
#define X_OFF   96
#define OUT_OFF 112
#define DUMMY_B (1228 * 4)   // byte offset (rel. to s_out) of the trash word

// ---------------------------------------------------------------------------
// Compile-time scatter table: for WMMA result w (w = 2*pair + kind; kind 0=re,
// 1=im), lane l, accumulator VGPR v, gives the byte offset inside the packed
// 1225-float output row (or DUMMY_B for masked elements).
// D layout of V_WMMA_F32_16X16X4_F32: VGPR v = row M=v (lanes 0-15) /
// M=v+8 (lanes 16-31), column N = lane & 15.
// ---------------------------------------------------------------------------
struct __align__(16) OffTab { unsigned short v[12][32][8]; };

constexpr OffTab make_tab() {
    OffTab t{};
    const int pti[6] = {0, 0, 0, 1, 1, 2};
    const int ptj[6] = {0, 1, 2, 1, 2, 2};
    for (int p = 0; p < 6; ++p) {
        const int ti = pti[p], tj = ptj[p];
        for (int lane = 0; lane < 32; ++lane) {
            const int jj = 16 * tj + (lane & 15);
            const int ib = 16 * ti + ((lane >> 4) << 3);
            for (int v = 0; v < 8; ++v) {
                const int ii = ib + v;
                int offRe = DUMMY_B, offIm = DUMMY_B;
                if (ii < 35 && jj < 35) {
                    if (ii <= jj)  // triu(35): 35*i - i(i-1)/2 + (j-i)
                        offRe = (ii * 35 - (ii * (ii - 1)) / 2 + (jj - ii)) * 4;
                    if (ii < jj)   // 630 + triu(35,1): 34*i - i(i-1)/2 + (j-i-1)
                        offIm = (630 + ii * 34 - (ii * (ii - 1)) / 2 + (jj - ii - 1)) * 4;
                }
                t.v[2 * p + 0][lane][v] = (unsigned short)offRe;
                t.v[2 * p + 1][lane][v] = (unsigned short)offIm;
            }
        }
    }
    return t;
}
__constant__ OffTab g_tab = make_tab();

__global__ __launch_bounds__(WAVES_PER_BLOCK * 32)
void biholo_k3_wmma_kernel(const float* __restrict__ xre,
                           const float* __restrict__ xim,
                           float* __restrict__ out, int nB) {
    __shared__ float smem[WAVES_PER_BLOCK * SLICE];
    const int lane = threadIdx.x & 31;
    const int wave = threadIdx.x >> 5;
    const int s = blockIdx.x * WAVES_PER_BLOCK + wave;
    if (s >= nB) return;

    float* sz_r  = &smem[wave * SLICE];
    float* sz_i  = sz_r + 48;
    float* sx    = sz_r + X_OFF;
    float* s_out = sz_r + OUT_OFF;
    char*  outb  = (char*)s_out;

    // Stage the 10 inputs into LDS so table-driven dynamic indexing stays in LDS.
    if (lane < 5) {
        sx[lane]     = xre[s * 5 + lane];
        sx[8 + lane] = xim[s * 5 + lane];
    }
    // Zero the padded monomial region.
    for (int m = lane; m < 48; m += 32) { sz_r[m] = 0.0f; sz_i[m] = 0.0f; }
    __builtin_amdgcn_wave_barrier();   // DS ops are in-order per wave

    // Each lane computes monomials m, m+32: zz = z[i]*z[j]*z[k].
    for (int m = lane; m < 35; m += 32) {
        const int i = c_TI[m], j = c_TJ[m], k = c_TK[m];
        const float air = sx[i],     aii = sx[8 + i];
        const float ajr = sx[j],     aji = sx[8 + j];
        const float akr = sx[k],     aki = sx[8 + k];
        const float pr = air * ajr - aii * aji;
        const float pi = air * aji + aii * ajr;
        sz_r[m] = pr * akr - pi * aki;
        sz_i[m] = pr * aki + pi * akr;
    }
    __builtin_amdgcn_wave_barrier();

    // f32 WMMA fragments for the three 16-wide tiles.
    // A 16x4: VGPR0 = K0 (lanes 0-15) / K2 (lanes 16-31); VGPR1 = K1 / K3.
    // B 4x16: VGPR0 = row K0 (lanes 0-15) / row K2 (lanes 16-31); VGPR1 = K1 / K3.
    // K = (re, im, 0, 0), so lanes 16-31 carry zeros.
    const int  lj = lane & 15;
    const bool lo = lane < 16;
    v2f Are[3], Aim[3], Bf[3];
#pragma unroll
    for (int t = 0; t < 3; ++t) {
        const float r0 = sz_r[16 * t + lj];
        const float i0 = sz_i[16 * t + lj];
        const float fr = lo ? r0 : 0.0f;
        const float fi = lo ? i0 : 0.0f;
        Are[t][0] = fr;  Are[t][1] = fi;    // dot -> Re_i*Re_j + Im_i*Im_j = Re(z_i conj(z_j))
        Aim[t][0] = fi;  Aim[t][1] = -fr;   // dot -> Im_i*Re_j - Re_i*Im_j = Im(z_i conj(z_j))
        Bf[t][0]  = fr;  Bf[t][1]  = fi;
    }

    // 6 upper-triangular tile pairs x {re, im} = 12 WMMAs; branch-free scatter
    // through the precomputed offset table (masked elements hit the dummy word).
#define SCAT2(pk, dd, v0)                                   \
    *(float*)(outb + ((pk) & 0xffffu)) = dd[(v0)];          \
    *(float*)(outb + ((pk) >> 16))     = dd[(v0) + 1];

    const int PT_I[6] = {0, 0, 0, 1, 1, 2};
    const int PT_J[6] = {0, 1, 2, 1, 2, 2};
#pragma unroll
    for (int p = 0; p < 6; ++p) {
        const int ti = PT_I[p], tj = PT_J[p];
        v8f cz = {};
        v8f dre = __builtin_amdgcn_wmma_f32_16x16x4_f32(
            false, Are[ti], false, Bf[tj], (short)0, cz, false, false);
        v8f dim = __builtin_amdgcn_wmma_f32_16x16x4_f32(
            false, Aim[ti], false, Bf[tj], (short)0, cz, false, false);

        const uint4 pr = *reinterpret_cast<const uint4*>(&g_tab.v[2 * p + 0][lane][0]);
        const uint4 pm = *reinterpret_cast<const uint4*>(&g_tab.v[2 * p + 1][lane][0]);

        SCAT2(pr.x, dre, 0) SCAT2(pr.y, dre, 2) SCAT2(pr.z, dre, 4) SCAT2(pr.w, dre, 6)
        SCAT2(pm.x, dim, 0) SCAT2(pm.y, dim, 2) SCAT2(pm.z, dim, 4) SCAT2(pm.w, dim, 6)
    }
#undef SCAT2
    __builtin_amdgcn_wave_barrier();

    // Coalesced writeback of the packed 1225-float row (row base is only
    // 4B-aligned since 1225*4 % 16 != 0, so b32 stores). 1225 = 38*32 + 9.
    float* __restrict__ orow = out + (size_t)s * 1225;
#pragma unroll
    for (int u = 0; u < 38; ++u) {
        const int c = lane + 32 * u;
        orow[c] = s_out[c];
    }
    if (lane < 9) {
        const int c = 1216 + lane;
        orow[c] = s_out[c];
    }
}

extern "C" void kernel_launch(void* const* d_in, const int* in_sizes, int n_in,
                              void* d_out, int out_size, void* d_ws, size_t ws_size,
                              hipStream_t stream) {
    const float* xre = (const float*)d_in[0];
    const float* xim = (const float*)d_in[1];
    float* out = (float*)d_out;
    const int nB = in_sizes[0] / 5;                 // (B, 5) float32
    const int blocks = (nB + WAVES_PER_BLOCK - 1) / WAVES_PER_BLOCK;
    biholo_k3_wmma_kernel<<<blocks, WAVES_PER_BLOCK * 32, 0, stream>>>(xre, xim, out, nB);
}